// SDRLayerDynamic_26482768347665
// MI455X (gfx1250) — compile-verified
//
#include <hip/hip_runtime.h>
#include <hip/hip_bf16.h>
#include <math.h>

#define NN    50000
#define EE    800000
#define RR    4
#define ETOT  (EE + NN)        // 850000 edges incl. self loops
#define NTILES (NN / 16)       // 3125 exact

typedef float v2f __attribute__((ext_vector_type(2)));
typedef float v8f __attribute__((ext_vector_type(8)));

__device__ __forceinline__ v8f wmma4(v2f a, v2f b, v8f c) {
  // V_WMMA_F32_16X16X4_F32 : D(16x16 f32) = A(16x4) * B(4x16) + C
  return __builtin_amdgcn_wmma_f32_16x16x4_f32(false, a, false, b, (short)0, c,
                                               false, false);
}

// float <-> order-preserving unsigned encoding (for atomic segment-max)
__device__ __forceinline__ unsigned f2ord(float f) {
  unsigned u = __float_as_uint(f);
  return (u & 0x80000000u) ? ~u : (u | 0x80000000u);
}
__device__ __forceinline__ float ord2f(unsigned u) {
  unsigned v = (u & 0x80000000u) ? (u ^ 0x80000000u) : ~u;
  return __uint_as_float(v);
}

// ---------------------------------------------------------------------------
// init: neigh = gat_bias broadcast, segment-max keys = ord(-inf), denom = 0
// ---------------------------------------------------------------------------
__global__ void init_kernel(const float* __restrict__ gbias,
                            float* __restrict__ neigh,
                            unsigned* __restrict__ mkey,
                            float* __restrict__ denom) {
  int i = blockIdx.x * 256 + threadIdx.x;
  if (i < RR * NN * 64) {
    int r = i / (NN * 64);
    int c = i & 63;
    neigh[i] = gbias[r * 64 + c];
  }
  if (i < RR * NN) {
    mkey[i]  = 0x007FFFFFu;   // ord(-inf)
    denom[i] = 0.0f;
  }
}

// ---------------------------------------------------------------------------
// h[r] = x @ W[r]   (WMMA f32 16x16x4; one wave = 16 rows x 64 cols)
// ---------------------------------------------------------------------------
__global__ void gemm_h_kernel(const float* __restrict__ x,
                              const float* __restrict__ W,
                              float* __restrict__ h) {
  const int r    = blockIdx.y;
  const int wave = threadIdx.x >> 5;
  const int lane = threadIdx.x & 31;
  const int tile = blockIdx.x * 4 + wave;
  if (tile >= NTILES) return;               // wave-uniform: EXEC stays all-ones
  const int hi   = lane >> 4;               // 0: K pair {0,1}, 1: K pair {2,3}
  const int lm   = lane & 15;
  const int row0 = tile * 16;

  const float* xrow = x + (size_t)(row0 + lm) * 64;
  v2f a[16];
#pragma unroll
  for (int kk = 0; kk < 16; ++kk) {
    a[kk][0] = xrow[kk * 4 + 2 * hi];
    a[kk][1] = xrow[kk * 4 + 2 * hi + 1];
  }

  const float* Wr = W + (size_t)r * 64 * 64;   // W[r][k][n]
  v8f acc0 = {}, acc1 = {}, acc2 = {}, acc3 = {};
#pragma unroll
  for (int kk = 0; kk < 16; ++kk) {
    const int k = kk * 4 + 2 * hi;
    const float* w0 = Wr + (size_t)k * 64;
    v2f b;
    b[0] = w0[lm];          b[1] = w0[64 + lm];          acc0 = wmma4(a[kk], b, acc0);
    b[0] = w0[16 + lm];     b[1] = w0[64 + 16 + lm];     acc1 = wmma4(a[kk], b, acc1);
    b[0] = w0[32 + lm];     b[1] = w0[64 + 32 + lm];     acc2 = wmma4(a[kk], b, acc2);
    b[0] = w0[48 + lm];     b[1] = w0[64 + 48 + lm];     acc3 = wmma4(a[kk], b, acc3);
  }

  float* hp = h + ((size_t)r * NN + row0) * 64;
#pragma unroll
  for (int j = 0; j < 8; ++j) {
    const int rr = j + 8 * hi;               // C layout: vgpr j -> rows j / j+8
    hp[(size_t)rr * 64 + lm]      = acc0[j];
    hp[(size_t)rr * 64 + 16 + lm] = acc1[j];
    hp[(size_t)rr * 64 + 32 + lm] = acc2[j];
    hp[(size_t)rr * 64 + 48 + lm] = acc3[j];
  }
}

// ---------------------------------------------------------------------------
// alpha_s / alpha_d per (relation, node)
// ---------------------------------------------------------------------------
__global__ void alpha_kernel(const float* __restrict__ h,
                             const float* __restrict__ att_s,
                             const float* __restrict__ att_d,
                             float* __restrict__ as_, float* __restrict__ ad_) {
  int i = blockIdx.x * 256 + threadIdx.x;    // i = r*NN + n
  if (i >= RR * NN) return;
  int r = i / NN;
  const float4* hp = (const float4*)(h + ((size_t)i << 6));
  const float4* sp = (const float4*)(att_s + r * 64);
  const float4* dp = (const float4*)(att_d + r * 64);
  float s = 0.f, d = 0.f;
#pragma unroll
  for (int k = 0; k < 16; ++k) {
    float4 hv = hp[k], sv = sp[k], dv = dp[k];
    s += hv.x * sv.x + hv.y * sv.y + hv.z * sv.z + hv.w * sv.w;
    d += hv.x * dv.x + hv.y * dv.y + hv.z * dv.z + hv.w * dv.w;
  }
  as_[i] = s;
  ad_[i] = d;
}

// ---------------------------------------------------------------------------
// pass 1: logits + atomic segment-max per destination
// ---------------------------------------------------------------------------
__global__ void edge_logit_max(const int* __restrict__ edges,
                               const float* __restrict__ as_,
                               const float* __restrict__ ad_,
                               float* __restrict__ elog,
                               unsigned* __restrict__ mkey) {
  const int r = blockIdx.y;
  const int e = blockIdx.x * 256 + threadIdx.x;
  if (e >= ETOT) return;
  int s, d;
  if (e < EE) {
    s = edges[((size_t)r * 2 + 0) * EE + e];
    d = edges[((size_t)r * 2 + 1) * EE + e];
  } else {
    s = d = e - EE;                          // self loop
  }
  float l = as_[r * NN + s] + ad_[r * NN + d];
  l = l > 0.f ? l : 0.2f * l;                // leaky_relu(0.2)
  elog[(size_t)r * ETOT + e] = l;
  atomicMax(&mkey[r * NN + d], f2ord(l));
}

// ---------------------------------------------------------------------------
// pass 2: e = exp(l - m[dst]); denom[dst] += e
// ---------------------------------------------------------------------------
__global__ void edge_exp_sum(const int* __restrict__ edges,
                             const unsigned* __restrict__ mkey,
                             float* __restrict__ elog,
                             float* __restrict__ denom) {
  const int r = blockIdx.y;
  const int e = blockIdx.x * 256 + threadIdx.x;
  if (e >= ETOT) return;
  int d = (e < EE) ? edges[((size_t)r * 2 + 1) * EE + e] : (e - EE);
  float m = ord2f(mkey[r * NN + d]);
  float w = __expf(elog[(size_t)r * ETOT + e] - m);
  elog[(size_t)r * ETOT + e] = w;
  atomicAdd(&denom[r * NN + d], w);
}

// ---------------------------------------------------------------------------
// pass 3: neigh[dst] += alpha * h[src]  (one 32-lane wave per edge, 2 cols/lane)
// ---------------------------------------------------------------------------
__global__ void edge_scatter(const int* __restrict__ edges,
                             const float* __restrict__ elog,
                             const float* __restrict__ denom,
                             const float* __restrict__ h,
                             float* __restrict__ neigh) {
  const int r    = blockIdx.y;
  const int lane = threadIdx.x & 31;
  const int e    = blockIdx.x * 8 + (threadIdx.x >> 5);
  if (e >= ETOT) return;
  int s, d;
  if (e < EE) {
    s = edges[((size_t)r * 2 + 0) * EE + e];
    d = edges[((size_t)r * 2 + 1) * EE + e];
  } else {
    s = d = e - EE;
  }
  float a = elog[(size_t)r * ETOT + e] / denom[r * NN + d];
  const float2 hv = *(const float2*)(h + (((size_t)r * NN + s) << 6) + lane * 2);
  float* np = neigh + (((size_t)r * NN + d) << 6) + lane * 2;
  atomicAdd(np,     hv.x * a);
  atomicAdd(np + 1, hv.y * a);
}

// ---------------------------------------------------------------------------
// fused MLP: out = tanh([x | neigh] @ w1 + b1) @ w2 + b2  (WMMA both GEMMs)
// ---------------------------------------------------------------------------
__global__ void mlp_kernel(const float* __restrict__ x,
                           const float* __restrict__ neigh,
                           const float* __restrict__ w1, const float* __restrict__ b1,
                           const float* __restrict__ w2, const float* __restrict__ b2,
                           float* __restrict__ out) {
  __shared__ float lds[4][16 * 68];          // per-wave 16x64 hdn tile, padded
  const int wave = threadIdx.x >> 5;
  const int lane = threadIdx.x & 31;
  const int tile = blockIdx.x * 4 + wave;
  if (tile >= NTILES) return;                // wave-uniform; no barriers used
  const int hi   = lane >> 4;
  const int lm   = lane & 15;
  const int row0 = tile * 16;

  // GEMM 1: K = 320 (64 from x, 4*64 from neigh)
  v8f acc[4] = {v8f{}, v8f{}, v8f{}, v8f{}};
  for (int kk = 0; kk < 80; ++kk) {
    const int k = kk * 4 + 2 * hi;           // column in combined
    const float* src;
    int kc;
    if (k < 64) { src = x + (size_t)(row0 + lm) * 64; kc = k; }
    else {
      const int r = (k - 64) >> 6;
      kc = (k - 64) & 63;
      src = neigh + ((size_t)r * NN + row0 + lm) * 64;
    }
    v2f a; a[0] = src[kc]; a[1] = src[kc + 1];
    const float* wrow = w1 + (size_t)k * 64;
#pragma unroll
    for (int t = 0; t < 4; ++t) {
      v2f b; b[0] = wrow[t * 16 + lm]; b[1] = wrow[64 + t * 16 + lm];
      acc[t] = wmma4(a, b, acc[t]);
    }
  }

  // bias + tanh, transpose via private LDS slab
  float* tl = lds[wave];
#pragma unroll
  for (int t = 0; t < 4; ++t) {
    const int col = t * 16 + lm;
    const float bias = b1[col];
#pragma unroll
    for (int j = 0; j < 8; ++j) {
      const int rr = j + 8 * hi;
      tl[rr * 68 + col] = tanhf(acc[t][j] + bias);
    }
  }

  // GEMM 2: K = 64 from LDS tile
  v8f o[4] = {v8f{}, v8f{}, v8f{}, v8f{}};
#pragma unroll
  for (int kk = 0; kk < 16; ++kk) {
    const int k = kk * 4 + 2 * hi;
    v2f a; a[0] = tl[lm * 68 + k]; a[1] = tl[lm * 68 + k + 1];
    const float* wrow = w2 + (size_t)k * 64;
#pragma unroll
    for (int t = 0; t < 4; ++t) {
      v2f b; b[0] = wrow[t * 16 + lm]; b[1] = wrow[64 + t * 16 + lm];
      o[t] = wmma4(a, b, o[t]);
    }
  }

  float* op = out + (size_t)row0 * 64;
#pragma unroll
  for (int t = 0; t < 4; ++t) {
    const int col = t * 16 + lm;
    const float bias = b2[col];
#pragma unroll
    for (int j = 0; j < 8; ++j) {
      const int rr = j + 8 * hi;
      op[(size_t)rr * 64 + col] = o[t][j] + bias;
    }
  }
}

// ---------------------------------------------------------------------------
extern "C" void kernel_launch(void* const* d_in, const int* in_sizes, int n_in,
                              void* d_out, int out_size, void* d_ws, size_t ws_size,
                              hipStream_t stream) {
  const float* x        = (const float*)d_in[0];
  const int*   edges    = (const int*)  d_in[1];
  const float* W        = (const float*)d_in[2];
  const float* att_src  = (const float*)d_in[3];
  const float* att_dst  = (const float*)d_in[4];
  const float* gat_bias = (const float*)d_in[5];
  const float* w1       = (const float*)d_in[6];
  const float* b1       = (const float*)d_in[7];
  const float* w2       = (const float*)d_in[8];
  const float* b2       = (const float*)d_in[9];
  float* out = (float*)d_out;

  // workspace layout (fp32 unless noted)
  float*    h       = (float*)d_ws;                          // R*N*64
  float*    neigh   = h + (size_t)RR * NN * 64;              // R*N*64
  float*    alpha_s = neigh + (size_t)RR * NN * 64;          // R*N
  float*    alpha_d = alpha_s + (size_t)RR * NN;             // R*N
  unsigned* mkey    = (unsigned*)(alpha_d + (size_t)RR * NN);// R*N (ordered-int)
  float*    denom   = (float*)(mkey + (size_t)RR * NN);      // R*N
  float*    elog    = denom + (size_t)RR * NN;               // R*(E+N)

  // 1. init neigh=bias, segment-max keys, denom
  init_kernel<<<dim3((RR * NN * 64 + 255) / 256), dim3(256), 0, stream>>>(
      gat_bias, neigh, mkey, denom);

  // 2. h[r] = x @ W[r]  (WMMA)
  gemm_h_kernel<<<dim3((NTILES + 3) / 4, RR), dim3(128), 0, stream>>>(x, W, h);

  // 3. attention coefficients
  alpha_kernel<<<dim3((RR * NN + 255) / 256), dim3(256), 0, stream>>>(
      h, att_src, att_dst, alpha_s, alpha_d);

  // 4-6. edge passes
  dim3 egrid((ETOT + 255) / 256, RR);
  edge_logit_max<<<egrid, dim3(256), 0, stream>>>(edges, alpha_s, alpha_d, elog, mkey);
  edge_exp_sum  <<<egrid, dim3(256), 0, stream>>>(edges, mkey, elog, denom);
  edge_scatter  <<<dim3((ETOT + 7) / 8, RR), dim3(256), 0, stream>>>(
      edges, elog, denom, h, neigh);

  // 7. fused MLP (WMMA x2)
  mlp_kernel<<<dim3((NTILES + 3) / 4), dim3(128), 0, stream>>>(
      x, neigh, w1, b1, w2, b2, out);
}